// Supporter2_91259465105799
// MI455X (gfx1250) — compile-verified
//
#include <hip/hip_runtime.h>
#include <hip/hip_bf16.h>

typedef __attribute__((ext_vector_type(2))) float v2f;
typedef __attribute__((ext_vector_type(8))) float v8f;

#define LRELU(v) ((v) > 0.0f ? (v) : 0.2f * (v))

// ---------------- degree / norm ----------------

__global__ void gcn_deg_init(float* __restrict__ deg, int n) {
    int i = blockIdx.x * blockDim.x + threadIdx.x;
    if (i < n) deg[i] = 1.0f;  // self loop
}

__global__ void gcn_deg_accum(const int* __restrict__ dst, float* __restrict__ deg, int e) {
    int i = blockIdx.x * blockDim.x + threadIdx.x;
    if (i < e) atomicAdd(&deg[dst[i]], 1.0f);
}

__global__ void gcn_dinv(float* __restrict__ deg, int n) {
    int i = blockIdx.x * blockDim.x + threadIdx.x;
    if (i < n) deg[i] = __frsqrt_rn(deg[i]);  // deg >= 1 always
}

// ---------------- GEMM1: h1[N,64] = x[N,128] @ W1[128,64] (f32 WMMA) ----------------
// One wave -> one 16x16 tile of h1. Block of 128 threads = 4 waves = the 4 N-tiles
// of one 16-row stripe, so A rows are loaded hot out of cache.

__global__ void __launch_bounds__(128) gcn_gemm1(const float* __restrict__ X,
                                                 const float* __restrict__ W1,
                                                 float* __restrict__ H1, int n) {
    const int lane  = threadIdx.x & 31;
    const int wave  = threadIdx.x >> 5;      // ntile 0..3
    const int m0    = blockIdx.x * 16;       // 16-row stripe
    const int n0    = wave * 16;             // 16-col stripe (of 64)
    const int l16   = lane & 15;
    const int kh    = lane >> 4;             // 0: K pair {0,1}; 1: K pair {2,3}

    int arow = m0 + l16;
    if (arow >= n) arow = n - 1;             // clamp loads; keep EXEC all-ones for WMMA
    const int bcol = n0 + l16;

    v8f c = {};
    const float* arow_p = X + (size_t)arow * 128;
#pragma unroll
    for (int k0 = 0; k0 < 128; k0 += 4) {
        v2f a, b;
        a.x = arow_p[k0 + 2 * kh + 0];
        a.y = arow_p[k0 + 2 * kh + 1];
        b.x = W1[(size_t)(k0 + 2 * kh + 0) * 64 + bcol];
        b.y = W1[(size_t)(k0 + 2 * kh + 1) * 64 + bcol];
        c = __builtin_amdgcn_wmma_f32_16x16x4_f32(false, a, false, b, (short)0, c, false, false);
    }

#pragma unroll
    for (int i = 0; i < 8; ++i) {
        int row = m0 + i + 8 * kh;           // VGPR i: M=i (lanes 0-15), M=i+8 (lanes 16-31)
        if (row < n) H1[(size_t)row * 64 + n0 + l16] = c[i];
    }
}

// ---------------- out1 init: bias + self-loop term ----------------

__global__ void gcn_out1_init(const float* __restrict__ H1, const float* __restrict__ dinv,
                              const float* __restrict__ b1, float* __restrict__ O1, int n) {
    int i = blockIdx.x * blockDim.x + threadIdx.x;
    if (i >= n * 64) return;
    int node = i >> 6, f = i & 63;
    float di = dinv[node];
    O1[i] = b1[f] + di * di * H1[i];
}

// ---------------- edge scatter, layer 1: 16 lanes per edge, float4 each ----------------

__global__ void gcn_scatter1(const int* __restrict__ src, const int* __restrict__ dst,
                             const float* __restrict__ H1, const float* __restrict__ dinv,
                             float* __restrict__ O1, int e) {
    long long t = (long long)blockIdx.x * blockDim.x + threadIdx.x;
    int edge = (int)(t >> 4);
    if (edge >= e) return;
    int f = (int)(t & 15) * 4;
    int s = src[edge], d = dst[edge];
    float norm = dinv[s] * dinv[d];
    const float4 hv = *(const float4*)(H1 + (size_t)s * 64 + f);
    float* o = O1 + (size_t)d * 64 + f;
    atomicAdd(o + 0, norm * hv.x);
    atomicAdd(o + 1, norm * hv.y);
    atomicAdd(o + 2, norm * hv.z);
    atomicAdd(o + 3, norm * hv.w);
}

// ---------------- GEMM2: hb[N,16] = leaky_relu(out1)[N,64] @ W2[64,16] (f32 WMMA) ----
// 4 waves per block handle 4 consecutive 16-row stripes (Fout = one 16-col tile).

__global__ void __launch_bounds__(128) gcn_gemm2(const float* __restrict__ O1,
                                                 const float* __restrict__ W2,
                                                 float* __restrict__ HB, int n) {
    const int lane = threadIdx.x & 31;
    const int wave = threadIdx.x >> 5;
    const int mtile = blockIdx.x * 4 + wave;
    const int ntiles = (n + 15) >> 4;
    if (mtile >= ntiles) return;
    const int m0 = mtile * 16;
    const int l16 = lane & 15;
    const int kh  = lane >> 4;

    int arow = m0 + l16;
    if (arow >= n) arow = n - 1;
    const float* arow_p = O1 + (size_t)arow * 64;

    v8f c = {};
#pragma unroll
    for (int k0 = 0; k0 < 64; k0 += 4) {
        v2f a, b;
        float a0 = arow_p[k0 + 2 * kh + 0];
        float a1 = arow_p[k0 + 2 * kh + 1];
        a.x = LRELU(a0);
        a.y = LRELU(a1);
        b.x = W2[(size_t)(k0 + 2 * kh + 0) * 16 + l16];
        b.y = W2[(size_t)(k0 + 2 * kh + 1) * 16 + l16];
        c = __builtin_amdgcn_wmma_f32_16x16x4_f32(false, a, false, b, (short)0, c, false, false);
    }

#pragma unroll
    for (int i = 0; i < 8; ++i) {
        int row = m0 + i + 8 * kh;
        if (row < n) HB[(size_t)row * 16 + l16] = c[i];
    }
}

// ---------------- out2 init (into d_out): bias + self-loop term ----------------

__global__ void gcn_out2_init(const float* __restrict__ HB, const float* __restrict__ dinv,
                              const float* __restrict__ b2, float* __restrict__ O2, int n) {
    int i = blockIdx.x * blockDim.x + threadIdx.x;
    if (i >= n * 16) return;
    int node = i >> 4, f = i & 15;
    float di = dinv[node];
    O2[i] = b2[f] + di * di * HB[i];
}

// ---------------- edge scatter, layer 2: 4 lanes per edge, float4 each ----------------

__global__ void gcn_scatter2(const int* __restrict__ src, const int* __restrict__ dst,
                             const float* __restrict__ HB, const float* __restrict__ dinv,
                             float* __restrict__ O2, int e) {
    long long t = (long long)blockIdx.x * blockDim.x + threadIdx.x;
    int edge = (int)(t >> 2);
    if (edge >= e) return;
    int f = (int)(t & 3) * 4;
    int s = src[edge], d = dst[edge];
    float norm = dinv[s] * dinv[d];
    const float4 hv = *(const float4*)(HB + (size_t)s * 16 + f);
    float* o = O2 + (size_t)d * 16 + f;
    atomicAdd(o + 0, norm * hv.x);
    atomicAdd(o + 1, norm * hv.y);
    atomicAdd(o + 2, norm * hv.z);
    atomicAdd(o + 3, norm * hv.w);
}

// ---------------- log_softmax over 16 columns, in place on d_out ----------------

__global__ void gcn_log_softmax(float* __restrict__ O2, int n) {
    int node = blockIdx.x * blockDim.x + threadIdx.x;
    if (node >= n) return;
    float v[16];
    float4* p = (float4*)(O2 + (size_t)node * 16);
#pragma unroll
    for (int q = 0; q < 4; ++q) {
        float4 t = p[q];
        v[4 * q + 0] = t.x; v[4 * q + 1] = t.y; v[4 * q + 2] = t.z; v[4 * q + 3] = t.w;
    }
    float m = v[0];
#pragma unroll
    for (int i = 1; i < 16; ++i) m = fmaxf(m, v[i]);
    float sum = 0.0f;
#pragma unroll
    for (int i = 0; i < 16; ++i) sum += __expf(v[i] - m);
    float lse = m + __logf(sum);
#pragma unroll
    for (int q = 0; q < 4; ++q) {
        float4 t;
        t.x = v[4 * q + 0] - lse; t.y = v[4 * q + 1] - lse;
        t.z = v[4 * q + 2] - lse; t.w = v[4 * q + 3] - lse;
        p[q] = t;
    }
}

extern "C" void kernel_launch(void* const* d_in, const int* in_sizes, int n_in,
                              void* d_out, int out_size, void* d_ws, size_t ws_size,
                              hipStream_t stream) {
    const float* x  = (const float*)d_in[0];
    const int*   ei = (const int*)d_in[1];
    const float* W1 = (const float*)d_in[2];
    const float* b1 = (const float*)d_in[3];
    const float* W2 = (const float*)d_in[4];
    const float* b2 = (const float*)d_in[5];

    const int n = in_sizes[0] / 128;   // nodes
    const int e = in_sizes[1] / 2;     // edges
    const int* src = ei;
    const int* dst = ei + e;

    // workspace layout (floats): dinv | h1 (reused as hb) | out1
    float* ws   = (float*)d_ws;
    float* dinv = ws;                                  // n
    float* h1   = ws + ((n + 4095) & ~4095);           // n*64 (later aliased by hb)
    float* out1 = h1 + (size_t)n * 64;                 // n*64
    float* hb   = h1;                                  // n*16, h1 is dead by then
    float* out2 = (float*)d_out;                       // n*16

    const int B = 256;

    gcn_deg_init<<<(n + B - 1) / B, B, 0, stream>>>(dinv, n);
    gcn_deg_accum<<<(e + B - 1) / B, B, 0, stream>>>(dst, dinv, e);
    gcn_dinv<<<(n + B - 1) / B, B, 0, stream>>>(dinv, n);

    gcn_gemm1<<<(n + 15) / 16, 128, 0, stream>>>(x, W1, h1, n);

    gcn_out1_init<<<((size_t)n * 64 + B - 1) / B, B, 0, stream>>>(h1, dinv, b1, out1, n);
    {
        long long work = (long long)e * 16;
        gcn_scatter1<<<(unsigned)((work + B - 1) / B), B, 0, stream>>>(src, dst, h1, dinv, out1, e);
    }

    gcn_gemm2<<<(((n + 15) / 16) + 3) / 4, 128, 0, stream>>>(out1, W2, hb, n);

    gcn_out2_init<<<((size_t)n * 16 + B - 1) / B, B, 0, stream>>>(hb, dinv, b2, out2, n);
    {
        long long work = (long long)e * 4;
        gcn_scatter2<<<(unsigned)((work + B - 1) / B), B, 0, stream>>>(src, dst, hb, dinv, out2, e);
    }

    gcn_log_softmax<<<(n + B - 1) / B, B, 0, stream>>>(out2, n);
}